// ST_GAT_70334384439384
// MI455X (gfx1250) — compile-verified
//
#include <hip/hip_runtime.h>
#include <hip/hip_bf16.h>
#include <math.h>

#define N_GRAPHS 32
#define N_NODES  2048
#define N_TOTAL  (N_GRAPHS * N_NODES)   // 65536
#define N_EDGES  1048576
#define HEADS    8
#define GAT_OUT  12
#define HID      (HEADS * GAT_OUT)      // 96
#define OUT_CH   9
#define T_SEQ    GAT_OUT                // 12
#define H1       32
#define H2       128
#define NEG_SLOPE 0.2f
#define BN_EPS    1e-5f

typedef float v2f __attribute__((ext_vector_type(2)));
typedef float v8f __attribute__((ext_vector_type(8)));

// ---- fp32 WMMA: D = A(16x4) * B(4x16) + C, full-precision accumulation ----
__device__ __forceinline__ v8f wmma_f32_16x16x4(v2f a, v2f b, v8f c) {
  // 8-arg pattern: (neg_a, A, neg_b, B, c_mod, C, reuse_a, reuse_b)
  return __builtin_amdgcn_wmma_f32_16x16x4_f32(false, a, false, b, (short)0, c,
                                               false, false);
}

// float atomic-max via monotonic int reinterpretation
__device__ __forceinline__ void atomicMaxF(float* addr, float val) {
  if (val >= 0.f) atomicMax((int*)addr, __float_as_int(val));
  else            atomicMin((unsigned int*)addr, __float_as_uint(val));
}

__device__ __forceinline__ float sigmoidf(float v) { return 1.f / (1.f + expf(-v)); }

// ---------------- init: zero accumulators, -inf the segment-max buffer ------
__global__ void k_init(float* acc, float* mbuf, float* denom) {
  int i = blockIdx.x * blockDim.x + threadIdx.x;
  if (i < N_TOTAL * HID)   acc[i]  = 0.f;
  if (i < N_TOTAL * HEADS) { mbuf[i] = -INFINITY; denom[i] = 0.f; }
}

// ---------------- GAT projection: xp = x @ W  (65536x12 @ 12x96), WMMA ------
__global__ void k_gat_project(const float* __restrict__ x,
                              const float* __restrict__ W,
                              float* __restrict__ xp) {
  int wave = blockIdx.x * (blockDim.x >> 5) + (threadIdx.x >> 5); // 4096 waves
  int lane = threadIdx.x & 31;
  int m0   = wave * 16;
  int hl   = lane & 15;
  int ksel = (lane >> 4) << 1;          // lanes 0-15: K=k0,k0+1; 16-31: k0+2,k0+3
  v2f a[3];
  int row = m0 + hl;
#pragma unroll
  for (int kt = 0; kt < 3; ++kt) {      // K = 12 = 3 x 4
    int kk = kt * 4 + ksel;
    a[kt].x = x[row * GAT_OUT + kk];
    a[kt].y = x[row * GAT_OUT + kk + 1];
  }
#pragma unroll
  for (int nt = 0; nt < 6; ++nt) {      // N = 96 = 6 x 16
    int n0 = nt * 16;
    v8f c = {};
#pragma unroll
    for (int kt = 0; kt < 3; ++kt) {
      int kk = kt * 4 + ksel;
      v2f b;
      b.x = W[kk * HID + n0 + hl];
      b.y = W[(kk + 1) * HID + n0 + hl];
      c = wmma_f32_16x16x4(a[kt], b, c);
    }
    int mbase = m0 + ((lane >> 4) << 3);
#pragma unroll
    for (int v = 0; v < 8; ++v)
      xp[(mbase + v) * HID + n0 + hl] = c[v];
  }
}

// ---------------- per-(node,head) attention coefficients --------------------
__global__ void k_attn_coeff(const float* __restrict__ xp,
                             const float* __restrict__ a_src,
                             const float* __restrict__ a_dst,
                             float* __restrict__ asrc, float* __restrict__ adst) {
  int i = blockIdx.x * blockDim.x + threadIdx.x;
  if (i >= N_TOTAL * HEADS) return;
  int h = i & 7, node = i >> 3;
  const float* p = xp + node * HID + h * GAT_OUT;
  float s = 0.f, d = 0.f;
#pragma unroll
  for (int c = 0; c < GAT_OUT; ++c) {
    s += p[c] * a_src[h * GAT_OUT + c];
    d += p[c] * a_dst[h * GAT_OUT + c];
  }
  asrc[i] = s; adst[i] = d;
}

__device__ __forceinline__ void edge_nodes(const long long* __restrict__ ei,
                                           int e, int& s, int& d) {
  if (e < N_EDGES) { s = (int)ei[e]; d = (int)ei[N_EDGES + e]; }
  else             { s = d = e - N_EDGES; }              // self loops
}

// ---------------- edge pass 1: segment max ----------------------------------
__global__ void k_edge_max(const long long* __restrict__ ei,
                           const float* __restrict__ asrc,
                           const float* __restrict__ adst,
                           float* __restrict__ mbuf) {
  int e = blockIdx.x * blockDim.x + threadIdx.x;
  if (e >= N_EDGES + N_TOTAL) return;
  int s, d; edge_nodes(ei, e, s, d);
#pragma unroll
  for (int h = 0; h < HEADS; ++h) {
    float v = asrc[s * HEADS + h] + adst[d * HEADS + h];
    v = v > 0.f ? v : NEG_SLOPE * v;
    atomicMaxF(&mbuf[d * HEADS + h], v);
  }
}

// ---------------- edge pass 2: exp & segment sum ----------------------------
__global__ void k_edge_sum(const long long* __restrict__ ei,
                           const float* __restrict__ asrc,
                           const float* __restrict__ adst,
                           const float* __restrict__ mbuf,
                           float* __restrict__ denom) {
  int e = blockIdx.x * blockDim.x + threadIdx.x;
  if (e >= N_EDGES + N_TOTAL) return;
  int s, d; edge_nodes(ei, e, s, d);
#pragma unroll
  for (int h = 0; h < HEADS; ++h) {
    float v = asrc[s * HEADS + h] + adst[d * HEADS + h];
    v = v > 0.f ? v : NEG_SLOPE * v;
    atomicAdd(&denom[d * HEADS + h], expf(v - mbuf[d * HEADS + h]));
  }
}

// ---------------- edge pass 3: alpha-weighted message scatter ---------------
__global__ void k_edge_msg(const long long* __restrict__ ei,
                           const float* __restrict__ asrc,
                           const float* __restrict__ adst,
                           const float* __restrict__ mbuf,
                           const float* __restrict__ denom,
                           const float* __restrict__ xp,
                           float* __restrict__ acc) {
  int e = blockIdx.x * blockDim.x + threadIdx.x;
  if (e >= N_EDGES + N_TOTAL) return;
  int s, d; edge_nodes(ei, e, s, d);
#pragma unroll
  for (int h = 0; h < HEADS; ++h) {
    float v = asrc[s * HEADS + h] + adst[d * HEADS + h];
    v = v > 0.f ? v : NEG_SLOPE * v;
    float w = expf(v - mbuf[d * HEADS + h]);
    float alpha = w / (denom[d * HEADS + h] + 1e-16f);
    const float* sf = xp + s * HID + h * GAT_OUT;
    float* da = acc + d * HID + h * GAT_OUT;
#pragma unroll
    for (int c = 0; c < GAT_OUT; ++c)
      atomicAdd(&da[c], sf[c] * alpha);
  }
}

// ---- head-mean + bias + BN + log_softmax, emit sequence layout (T,B,F) -----
__global__ void k_finalize(const float* __restrict__ acc,
                           const float* __restrict__ gat_bias,
                           const float* __restrict__ gamma,
                           const float* __restrict__ beta,
                           const float* __restrict__ mean,
                           const float* __restrict__ var,
                           float* __restrict__ seq) {
  int n = blockIdx.x * blockDim.x + threadIdx.x;
  if (n >= N_TOTAL) return;
  float v[GAT_OUT];
  float mx = -INFINITY;
#pragma unroll
  for (int c = 0; c < GAT_OUT; ++c) {
    float s = 0.f;
#pragma unroll
    for (int h = 0; h < HEADS; ++h) s += acc[n * HID + h * GAT_OUT + c];
    s = s * (1.f / HEADS) + gat_bias[c];
    s = (s - mean[c]) * rsqrtf(var[c] + BN_EPS) * gamma[c] + beta[c];
    v[c] = s; mx = fmaxf(mx, s);
  }
  float lse = 0.f;
#pragma unroll
  for (int c = 0; c < GAT_OUT; ++c) lse += expf(v[c] - mx);
  lse = logf(lse) + mx;
  int g = n >> 11, nn = n & 2047;
#pragma unroll
  for (int c = 0; c < GAT_OUT; ++c)
    seq[(c * N_GRAPHS + g) * N_NODES + nn] = v[c] - lse;
}

// ---- hoisted LSTM1 input projection: (384,2048) @ (2048,128), WMMA ---------
__global__ void k_gemm_g1(const float* __restrict__ seq,
                          const float* __restrict__ w_ih1,
                          float* __restrict__ G1) {
  int wave = blockIdx.x * (blockDim.x >> 5) + (threadIdx.x >> 5); // 192 waves
  int lane = threadIdx.x & 31;
  int mt = wave >> 3, nt = wave & 7;
  int m0 = mt * 16, n0 = nt * 16;
  int hl = lane & 15, ksel = (lane >> 4) << 1;
  int arow = m0 + hl;
  int bn   = n0 + hl;
  v8f c = {};
  for (int k0 = 0; k0 < N_NODES; k0 += 4) {
    int kk = k0 + ksel;
    v2f a, b;
    a.x = seq[arow * N_NODES + kk];
    a.y = seq[arow * N_NODES + kk + 1];
    b.x = w_ih1[bn * N_NODES + kk];      // B[k][n] = w_ih1[n][k]
    b.y = w_ih1[bn * N_NODES + kk + 1];
    c = wmma_f32_16x16x4(a, b, c);
  }
  int mbase = m0 + ((lane >> 4) << 3);
#pragma unroll
  for (int v = 0; v < 8; ++v)
    G1[(mbase + v) * (4 * H1) + n0 + hl] = c[v];
}

// ---- fused sequential LSTM1+LSTM2 recurrence, state in LDS -----------------
__global__ void __launch_bounds__(512)
k_lstm(const float* __restrict__ G1,
       const float* __restrict__ w_hh1, const float* __restrict__ b_ih1,
       const float* __restrict__ b_hh1,
       const float* __restrict__ w_ih2, const float* __restrict__ w_hh2,
       const float* __restrict__ b_ih2, const float* __restrict__ b_hh2,
       float* __restrict__ h2_all) {
  __shared__ float h1[N_GRAPHS * H1], c1[N_GRAPHS * H1];
  __shared__ float h2[N_GRAPHS * H2], c2[N_GRAPHS * H2];
  int tid = threadIdx.x;
  for (int i = tid; i < N_GRAPHS * H1; i += blockDim.x) { h1[i] = 0.f; c1[i] = 0.f; }
  for (int i = tid; i < N_GRAPHS * H2; i += blockDim.x) { h2[i] = 0.f; c2[i] = 0.f; }
  __syncthreads();
  for (int t = 0; t < T_SEQ; ++t) {
    // LSTM1: 32x32 cells, 2 per thread
    float nh[2], nc[2];
#pragma unroll
    for (int r = 0; r < 2; ++r) {
      int o = tid + r * 512;
      int b = o >> 5, j = o & 31;
      float g4[4];
#pragma unroll
      for (int q = 0; q < 4; ++q) {
        int jj = q * H1 + j;               // gate order i,f,g,o
        float s = G1[(t * N_GRAPHS + b) * (4 * H1) + jj] + b_ih1[jj] + b_hh1[jj];
        const float* wr = w_hh1 + jj * H1;
        for (int k = 0; k < H1; ++k) s += h1[b * H1 + k] * wr[k];
        g4[q] = s;
      }
      float cc = sigmoidf(g4[1]) * c1[o] + sigmoidf(g4[0]) * tanhf(g4[2]);
      nc[r] = cc; nh[r] = sigmoidf(g4[3]) * tanhf(cc);
    }
    __syncthreads();
#pragma unroll
    for (int r = 0; r < 2; ++r) { int o = tid + r * 512; c1[o] = nc[r]; h1[o] = nh[r]; }
    __syncthreads();
    // LSTM2: 32x128 cells, 8 per thread
    float nh2[8], nc2[8];
#pragma unroll
    for (int r = 0; r < 8; ++r) {
      int o = tid + r * 512;
      int b = o >> 7, j = o & 127;
      float g4[4];
#pragma unroll
      for (int q = 0; q < 4; ++q) {
        int jj = q * H2 + j;
        float s = b_ih2[jj] + b_hh2[jj];
        const float* wi = w_ih2 + jj * H1;
        for (int k = 0; k < H1; ++k) s += h1[b * H1 + k] * wi[k];
        const float* wh = w_hh2 + jj * H2;
        for (int k = 0; k < H2; ++k) s += h2[b * H2 + k] * wh[k];
        g4[q] = s;
      }
      float cc = sigmoidf(g4[1]) * c2[o] + sigmoidf(g4[0]) * tanhf(g4[2]);
      nc2[r] = cc; nh2[r] = sigmoidf(g4[3]) * tanhf(cc);
    }
    __syncthreads();
#pragma unroll
    for (int r = 0; r < 8; ++r) {
      int o = tid + r * 512;
      c2[o] = nc2[r]; h2[o] = nh2[r];
      h2_all[t * N_GRAPHS * H2 + o] = nh2[r];
    }
    __syncthreads();
  }
}

// ---- final linear (last 9 steps): (288,128) @ (128,2048), WMMA -------------
__global__ void k_gemm_out(const float* __restrict__ h2_all,
                           const float* __restrict__ lin_w,
                           const float* __restrict__ lin_b,
                           float* __restrict__ out) {
  int wave = blockIdx.x * (blockDim.x >> 5) + (threadIdx.x >> 5); // 2304 waves
  int lane = threadIdx.x & 31;
  int mt = wave >> 7, nt = wave & 127;
  int m0 = mt * 16, n0 = nt * 16;
  int hl = lane & 15, ksel = (lane >> 4) << 1;
  int arow = m0 + 96 + hl;         // skip first 3 timesteps (row r -> h2 row r+96)
  int bn   = n0 + hl;
  v8f c = {};
  for (int k0 = 0; k0 < H2; k0 += 4) {
    int kk = k0 + ksel;
    v2f a, b;
    a.x = h2_all[arow * H2 + kk];
    a.y = h2_all[arow * H2 + kk + 1];
    b.x = lin_w[bn * H2 + kk];       // B[k][n] = lin_w[n][k]
    b.y = lin_w[bn * H2 + kk + 1];
    c = wmma_f32_16x16x4(a, b, c);
  }
  int mbase = m0 + ((lane >> 4) << 3);
  float bias = lin_b[n0 + hl];
#pragma unroll
  for (int v = 0; v < 8; ++v) {
    int r = mbase + v;               // 0..287
    int t = r >> 5, g = r & 31;      // t in 0..8 (last-9 index), g = graph
    out[(g * N_NODES + n0 + hl) * OUT_CH + t] = c[v] + bias;
  }
}

extern "C" void kernel_launch(void* const* d_in, const int* in_sizes, int n_in,
                              void* d_out, int out_size, void* d_ws, size_t ws_size,
                              hipStream_t stream) {
  const float*     x        = (const float*)d_in[0];
  const long long* ei       = (const long long*)d_in[1];
  const float*     W_gat    = (const float*)d_in[2];
  const float*     a_src    = (const float*)d_in[3];
  const float*     a_dst    = (const float*)d_in[4];
  const float*     gat_bias = (const float*)d_in[5];
  const float*     bn_gamma = (const float*)d_in[6];
  const float*     bn_beta  = (const float*)d_in[7];
  const float*     bn_mean  = (const float*)d_in[8];
  const float*     bn_var   = (const float*)d_in[9];
  const float*     w_ih1    = (const float*)d_in[10];
  const float*     w_hh1    = (const float*)d_in[11];
  const float*     b_ih1    = (const float*)d_in[12];
  const float*     b_hh1    = (const float*)d_in[13];
  const float*     w_ih2    = (const float*)d_in[14];
  const float*     w_hh2    = (const float*)d_in[15];
  const float*     b_ih2    = (const float*)d_in[16];
  const float*     b_hh2    = (const float*)d_in[17];
  const float*     lin_w    = (const float*)d_in[18];
  const float*     lin_b    = (const float*)d_in[19];

  float* ws    = (float*)d_ws;
  float* xp    = ws; ws += (size_t)N_TOTAL * HID;     // 25.2 MB
  float* acc   = ws; ws += (size_t)N_TOTAL * HID;     // 25.2 MB
  float* asrc  = ws; ws += (size_t)N_TOTAL * HEADS;
  float* adst  = ws; ws += (size_t)N_TOTAL * HEADS;
  float* mbuf  = ws; ws += (size_t)N_TOTAL * HEADS;
  float* denom = ws; ws += (size_t)N_TOTAL * HEADS;
  float* seq   = ws; ws += (size_t)T_SEQ * N_TOTAL;
  float* G1    = ws; ws += (size_t)T_SEQ * N_GRAPHS * 4 * H1;
  float* h2a   = ws; ws += (size_t)T_SEQ * N_GRAPHS * H2;

  int TE = N_EDGES + N_TOTAL;
  k_init<<<(N_TOTAL * HID + 255) / 256, 256, 0, stream>>>(acc, mbuf, denom);
  k_gat_project<<<512, 256, 0, stream>>>(x, W_gat, xp);
  k_attn_coeff<<<(N_TOTAL * HEADS + 255) / 256, 256, 0, stream>>>(xp, a_src, a_dst, asrc, adst);
  k_edge_max<<<(TE + 255) / 256, 256, 0, stream>>>(ei, asrc, adst, mbuf);
  k_edge_sum<<<(TE + 255) / 256, 256, 0, stream>>>(ei, asrc, adst, mbuf, denom);
  k_edge_msg<<<(TE + 255) / 256, 256, 0, stream>>>(ei, asrc, adst, mbuf, denom, xp, acc);
  k_finalize<<<(N_TOTAL + 255) / 256, 256, 0, stream>>>(acc, gat_bias, bn_gamma, bn_beta,
                                                        bn_mean, bn_var, seq);
  k_gemm_g1<<<24, 256, 0, stream>>>(seq, w_ih1, G1);
  k_lstm<<<1, 512, 0, stream>>>(G1, w_hh1, b_ih1, b_hh1, w_ih2, w_hh2, b_ih2, b_hh2, h2a);
  k_gemm_out<<<288, 256, 0, stream>>>(h2a, lin_w, lin_b, (float*)d_out);
}